// Huber_40252433498727
// MI455X (gfx1250) — compile-verified
//
#include <hip/hip_runtime.h>
#include <hip/hip_bf16.h>

typedef float v2f __attribute__((ext_vector_type(2)));
typedef float v8f __attribute__((ext_vector_type(8)));

#define HC     1.345f
#define HGAMMA 2.0f
#define HALPHA 0.355084f

// ---------------------------------------------------------------- gram 20x20
__global__ __launch_bounds__(512) void gram20(const float* __restrict__ M,
                                              int len, int transposed,
                                              float* __restrict__ G) {
  int t = threadIdx.x;
  if (t >= 400) return;
  int p = t / 20, q = t % 20;
  float acc = 0.f;
  if (transposed) {  // M [20,len]: elem(i,p)=M[p*len+i]
    const float* rp = M + (size_t)p * len;
    const float* rq = M + (size_t)q * len;
    for (int i = 0; i < len; ++i) acc += rp[i] * rq[i];
  } else {           // M [len,20]
    for (int i = 0; i < len; ++i) acc += M[i * 20 + p] * M[i * 20 + q];
  }
  G[t] = acc;
}

// ------------------------------------------------- 20x20 Gauss-Jordan invert
__global__ __launch_bounds__(64) void invert20(const float* __restrict__ G,
                                               float* __restrict__ Gi) {
  __shared__ float A[20][40];
  __shared__ float fac[20];
  __shared__ int pivR;
  int t = threadIdx.x;
  for (int idx = t; idx < 800; idx += 64) {
    int r = idx / 40, c = idx % 40;
    A[r][c] = (c < 20) ? G[r * 20 + c] : ((c - 20) == r ? 1.f : 0.f);
  }
  __syncthreads();
  for (int k = 0; k < 20; ++k) {
    if (t == 0) {
      int best = k; float bv = fabsf(A[k][k]);
      for (int r = k + 1; r < 20; ++r) {
        float v = fabsf(A[r][k]);
        if (v > bv) { bv = v; best = r; }
      }
      pivR = best;
    }
    __syncthreads();
    int pr = pivR;
    if (pr != k)
      for (int c = t; c < 40; c += 64) {
        float tmp = A[k][c]; A[k][c] = A[pr][c]; A[pr][c] = tmp;
      }
    __syncthreads();
    float piv = A[k][k];
    __syncthreads();
    for (int c = t; c < 40; c += 64) A[k][c] = A[k][c] / piv;
    __syncthreads();
    if (t < 20) fac[t] = A[t][k];
    __syncthreads();
    for (int idx = t; idx < 800; idx += 64) {
      int r = idx / 40, c = idx % 40;
      if (r != k) A[r][c] -= fac[r] * A[k][c];
    }
    __syncthreads();
  }
  for (int idx = t; idx < 400; idx += 64)
    Gi[idx] = A[idx / 20][idx % 20 + 20];
}

// ------------------------------------------------- pinv projection  W = Gi*M^T
__global__ __launch_bounds__(256) void proj20(const float* __restrict__ Gi,
                                              const float* __restrict__ M,
                                              float* __restrict__ out,
                                              int len, int transposed) {
  int idx = blockIdx.x * blockDim.x + threadIdx.x;
  if (idx >= 20 * len) return;
  int p = idx / len, i = idx % len;
  float acc = 0.f;
  if (transposed) {  // M [20,len] : out[p,i] = sum_k Gi[p,k]*M[k,i]
    for (int k = 0; k < 20; ++k) acc += Gi[p * 20 + k] * M[k * len + i];
  } else {           // M [len,20] : out[p,i] = sum_k Gi[p,k]*M[i,k]
    for (int k = 0; k < 20; ++k) acc += Gi[p * 20 + k] * M[i * 20 + k];
  }
  out[(size_t)p * len + i] = acc;
}

// ---------------------------------------------------------------- bitonic 2048
__device__ __forceinline__ void bitonic2048(float* s, int tid) {
  for (int k = 2; k <= 2048; k <<= 1)
    for (int j = k >> 1; j > 0; j >>= 1) {
      __syncthreads();
      for (int i = tid; i < 2048; i += 256) {
        int ixj = i ^ j;
        if (ixj > i) {
          float a = s[i], b = s[ixj];
          bool up = ((i & k) == 0);
          if ((a > b) == up) { s[i] = b; s[ixj] = a; }
        }
      }
    }
  __syncthreads();
}

// --------------------------------------------------------- batched Huber IRLS
// One block = 16 independent regressions (columns of X or rows of X).
//   X   [1600,1200]
//   Xm  : colPhase ? U [1600,20]      : Vnew [20,1200]   (design matrix access)
//   W   : colPhase ? Up [20,1600]     : VpT  [20,1200]   (pinv, shared)
//   betaIn : colPhase ? V [20,1200]   : U [1600,20]
//   kers   : pre-offset kernel base, 9 floats per problem
__global__ __launch_bounds__(256) void hubreg_batch(
    const float* __restrict__ X, const float* __restrict__ Xm,
    const float* __restrict__ W, const float* __restrict__ betaIn,
    const float* __restrict__ kers, float* __restrict__ betaOut,
    int Nlen, int colPhase) {
  const int n = 1200;
  const int tid  = threadIdx.x;
  const int wave = tid >> 5;
  const int lane = tid & 31;
  const int half = lane >> 4;
  const int lm   = lane & 15;
  const int b0   = blockIdx.x * 16;
  const int nCh  = Nlen >> 4;           // 100 or 75 (exact)
  const int medIdx = (Nlen - 1) >> 1;   // 799 or 599
  const float ndf  = (float)(Nlen - 21);

  __shared__ float sbuf[2048];
  __shared__ float rcol[1600];
  __shared__ float betaS[20][16];
  __shared__ float kerL[16][9];
  __shared__ float scaleS[16];
  __shared__ float Ssum[16];
  __shared__ float ztile[8][16][17];
  __shared__ float Tacc[3][32][16];

  auto Xel = [&](int row, int c) -> float {
    return colPhase ? X[(size_t)row * n + b0 + c] : X[(size_t)(b0 + c) * n + row];
  };
  auto Ael = [&](int row, int p) -> float {
    return colPhase ? Xm[row * 20 + p] : Xm[(size_t)p * n + row];
  };

  for (int t = tid; t < 320; t += 256) {
    int p = t / 16, c = t % 16;
    betaS[p][c] = colPhase ? betaIn[(size_t)p * n + b0 + c]
                           : betaIn[(b0 + c) * 20 + p];
  }
  for (int t = tid; t < 144; t += 256) {
    int c = t / 9, k = t % 9;
    kerL[c][k] = kers[(b0 + c) * 9 + k];
  }
  __syncthreads();

  // ---- initial scale: 1.4815 * median(|r - median(r)|), lower-middle median
  for (int c = 0; c < 16; ++c) {
    for (int i = tid; i < 2048; i += 256) {
      float v;
      if (i < Nlen) {
        float acc = Xel(i, c);
#pragma unroll
        for (int p = 0; p < 20; ++p) acc -= Ael(i, p) * betaS[p][c];
        rcol[i] = acc; v = acc;
      } else v = __builtin_inff();
      sbuf[i] = v;
    }
    __syncthreads();
    bitonic2048(sbuf, tid);
    float med1 = sbuf[medIdx];
    __syncthreads();
    for (int i = tid; i < 2048; i += 256)
      sbuf[i] = (i < Nlen) ? fabsf(rcol[i] - med1) : __builtin_inff();
    __syncthreads();
    bitonic2048(sbuf, tid);
    if (tid == 0) scaleS[c] = 1.4815f * sbuf[medIdx];
    __syncthreads();
  }

  // residual 16x16 tile via V_WMMA_F32_16X16X4_F32 :  r = X_tile - Xm*beta
  auto rtile = [&](int ch) -> v8f {
    v8f acc;
#pragma unroll
    for (int v = 0; v < 8; ++v) acc[v] = Xel(ch * 16 + v + 8 * half, lm);
#pragma unroll
    for (int kk = 0; kk < 5; ++kk) {
      int p0 = 4 * kk + 2 * half;
      int row = ch * 16 + lm;
      v2f a, b;
      a[0] = Ael(row, p0);
      a[1] = Ael(row, p0 + 1);
      b[0] = -betaS[p0][lm];       // negate B: f32 WMMA has no A/B NEG
      b[1] = -betaS[p0 + 1][lm];
      acc = __builtin_amdgcn_wmma_f32_16x16x4_f32(false, a, false, b,
                                                  (short)0, acc, false, false);
    }
    return acc;
  };

  // ---------------------------------------------------------- 3 IRLS iters
  for (int it = 0; it < 3; ++it) {
    for (int t = tid; t < 16; t += 256) Ssum[t] = 0.f;
    for (int t = tid; t < 3 * 32 * 16; t += 256) (&Tacc[0][0][0])[t] = 0.f;
    __syncthreads();

    // pass 1: S[c] = sum_i 0.5*min((r/scale)^2, C^2)
    for (int ch = wave; ch < nCh; ch += 8) {
      v8f r = rtile(ch);
      float inv = 1.f / scaleS[lm];
      float part = 0.f;
#pragma unroll
      for (int v = 0; v < 8; ++v) {
        float t2 = r[v] * inv; t2 *= t2;
        part += 0.5f * fminf(t2, HC * HC);
      }
      part += __shfl_xor(part, 16, 32);
      if (lane < 16) atomicAdd(&Ssum[lane], part);
    }
    __syncthreads();
    if (tid < 16) {
      float sc = scaleS[tid];
      scaleS[tid] = sqrtf(HGAMMA * sc * sc / (2.f * HALPHA * ndf) * Ssum[tid]);
    }
    __syncthreads();

    // pass 2: T[s][pp][c] = sum_q W[pp, q+s-1] * clamp(r, +-C*scale)[q,c]
    v8f tAcc[2][3];
#pragma unroll
    for (int mt = 0; mt < 2; ++mt)
#pragma unroll
      for (int s = 0; s < 3; ++s)
#pragma unroll
        for (int v = 0; v < 8; ++v) tAcc[mt][s][v] = 0.f;

    for (int ch = wave; ch < nCh; ch += 8) {
      v8f r = rtile(ch);
      float lim = HC * scaleS[lm];
#pragma unroll
      for (int v = 0; v < 8; ++v)
        ztile[wave][v + 8 * half][lm] = fminf(fmaxf(r[v], -lim), lim);
      __builtin_amdgcn_wave_barrier();
#pragma unroll
      for (int kk = 0; kk < 4; ++kk) {
        v2f bf;
        bf[0] = ztile[wave][4 * kk + 2 * half][lm];
        bf[1] = ztile[wave][4 * kk + 2 * half + 1][lm];
#pragma unroll
        for (int mt = 0; mt < 2; ++mt) {
          int pp = mt * 16 + lm;
          bool ppok = pp < 20;
          const float* wrow = W + (size_t)pp * Nlen;
#pragma unroll
          for (int s = 0; s < 3; ++s) {
            int q0 = ch * 16 + 4 * kk + 2 * half + s - 1;
            int q1 = q0 + 1;
            v2f af;
            af[0] = (ppok && q0 >= 0 && q0 < Nlen) ? wrow[q0] : 0.f;
            af[1] = (ppok && q1 >= 0 && q1 < Nlen) ? wrow[q1] : 0.f;
            tAcc[mt][s] = __builtin_amdgcn_wmma_f32_16x16x4_f32(
                false, af, false, bf, (short)0, tAcc[mt][s], false, false);
          }
        }
      }
    }
#pragma unroll
    for (int mt = 0; mt < 2; ++mt)
#pragma unroll
      for (int s = 0; s < 3; ++s)
#pragma unroll
        for (int v = 0; v < 8; ++v)
          atomicAdd(&Tacc[s][mt * 16 + v + 8 * half][lm], tAcc[mt][s][v]);
    __syncthreads();

    // delta[p][c] = sum_{u,s} kerEff[u][s] * T[s][p+u-1][c] ; beta += delta
    for (int t = tid; t < 320; t += 256) {
      int p = t / 16, c = t % 16;
      float d = 0.f;
#pragma unroll
      for (int u = 0; u < 3; ++u) {
        int pp = p + u - 1;
        if (pp < 0) continue;
#pragma unroll
        for (int s = 0; s < 3; ++s) {
          float kv = colPhase ? kerL[c][u * 3 + s] : kerL[c][s * 3 + u];
          d += kv * Tacc[s][pp][c];
        }
      }
      betaS[p][c] += d;
    }
    __syncthreads();
  }

  for (int t = tid; t < 320; t += 256) {
    int p = t / 16, c = t % 16;
    if (colPhase) betaOut[(size_t)p * n + b0 + c] = betaS[p][c];
    else          betaOut[(b0 + c) * 20 + p]      = betaS[p][c];
  }
}

// ---------------------------------------------------------------------------
extern "C" void kernel_launch(void* const* d_in, const int* in_sizes, int n_in,
                              void* d_out, int out_size, void* d_ws,
                              size_t ws_size, hipStream_t stream) {
  (void)in_sizes; (void)n_in; (void)out_size; (void)ws_size;
  const int m = 1600, n = 1200, r = 20;
  const float* X  = (const float*)d_in[0];
  const float* U  = (const float*)d_in[1];
  const float* V  = (const float*)d_in[2];
  const float* Kn = (const float*)d_in[3];
  float* out  = (float*)d_out;
  float* Xout = out;                       // [1600,1200]
  float* Unew = out + (size_t)m * n;       // [1600,20]
  float* Vnew = out + (size_t)m * n + (size_t)m * r;  // [20,1200]

  float* ws  = (float*)d_ws;
  float* Up  = ws;            // 20*1600
  float* VpT = ws + 32000;    // 20*1200
  float* G   = ws + 56000;
  float* Gi  = ws + 56400;
  float* G2  = ws + 56800;
  float* G2i = ws + 57200;

  hipMemcpyAsync(Xout, X, (size_t)m * n * sizeof(float),
                 hipMemcpyDeviceToDevice, stream);

  // ---- column phase: Up = pinv(U) = inv(U^T U) U^T
  gram20<<<1, 512, 0, stream>>>(U, m, 0, G);
  invert20<<<1, 64, 0, stream>>>(G, Gi);
  proj20<<<(20 * m + 255) / 256, 256, 0, stream>>>(Gi, U, Up, m, 0);
  hubreg_batch<<<n / 16, 256, 0, stream>>>(X, U, Up, V, Kn, Vnew, m, 1);

  // ---- row phase: Vp^T = inv(Vnew Vnew^T) Vnew
  gram20<<<1, 512, 0, stream>>>(Vnew, n, 1, G2);
  invert20<<<1, 64, 0, stream>>>(G2, G2i);
  proj20<<<(20 * n + 255) / 256, 256, 0, stream>>>(G2i, Vnew, VpT, n, 1);
  hubreg_batch<<<m / 16, 256, 0, stream>>>(X, Vnew, VpT, U, Kn + (n - 1) * 9,
                                           Unew, n, 0);
}